// BayesStateEstimator_31851477467822
// MI455X (gfx1250) — compile-verified
//
#include <hip/hip_runtime.h>

// ---------------------------------------------------------------------------
// LGSSM predict scan for MI455X (gfx1250, wave32, WMMA).
//   Kernel 1: serial scan  x_t = F x + u ; P_t = F P F^T + Q     (64 waves)
//   Kernel 2: parallel emit y = H x + d ; S = H P H^T + R        (131072 waves)
// Matrix products use V_WMMA_F32_16X16X4_F32 (D = A(16x4) * B(4x16) + C).
// ---------------------------------------------------------------------------

typedef __attribute__((ext_vector_type(2))) float v2f;
typedef __attribute__((ext_vector_type(8))) float v8f;

#define BS_  64
#define SEQ_ 2048
#define XD_  16
#define YD_  8
#define LP   18   // padded LDS row stride: conflict-free + 8B-aligned float2 reads

#if defined(__HIP_DEVICE_COMPILE__) && !__has_builtin(__builtin_amdgcn_wmma_f32_16x16x4_f32)
#error "missing __builtin_amdgcn_wmma_f32_16x16x4_f32 on this toolchain (device pass)"
#endif

static __device__ __forceinline__ v8f wmma4(v2f a, v2f b, v8f c) {
#if defined(__HIP_DEVICE_COMPILE__)
  // 8-arg form: (neg_a, A, neg_b, B, c_mod, C, reuse_a, reuse_b)
  return __builtin_amdgcn_wmma_f32_16x16x4_f32(false, a, false, b, (short)0, c,
                                               false, false);
#else
  (void)a; (void)b;
  return c;  // host pass never executes device code; keep parser happy
#endif
}

// ---------------------------------------------------------------------------
// Kernel 1: one wave32 per batch element; strictly serial over t.
// Critical path per step: 2x(2 chained WMMA)+add, one LDS round trip of T,
// 2x(2 chained WMMA)+add. F/Q/u for t+1 are register-prefetched during step t.
// ---------------------------------------------------------------------------
__global__ __launch_bounds__(32, 1) void lgssm_scan(
    const float* __restrict__ x0, const float* __restrict__ P0,
    const float* __restrict__ Fs, const float* __restrict__ Qs,
    const float* __restrict__ us,
    float* __restrict__ xs, float* __restrict__ Ps)
{
  __shared__ float Pl[XD_ * LP];  // P_{t-1}, row-padded
  __shared__ float Tl[XD_ * LP];  // T = F*P intermediate
  __shared__ float xl[XD_];       // x_{t-1}

  const int b    = blockIdx.x;
  const int lane = threadIdx.x;
  const int lo   = lane & 15;
  const int hi   = lane >> 4;

  const float* __restrict__ Fb = Fs + (size_t)b * SEQ_ * 256;
  const float* __restrict__ Qb = Qs + (size_t)b * SEQ_ * 256;
  const float* __restrict__ ub = us + (size_t)b * SEQ_ * 16;
  float* __restrict__ Pob = Ps + (size_t)b * SEQ_ * 256;
  float* __restrict__ xob = xs + (size_t)b * SEQ_ * 16;

  // ---- initial state into LDS (C-layout rows i and i+8 across lane halves)
#pragma unroll
  for (int i = 0; i < 8; ++i) {
    const int m = i + 8 * hi;
    Pl[m * LP + lo] = P0[(size_t)b * 256 + m * 16 + lo];
  }
  if (lane < 16) xl[lane] = x0[(size_t)b * 16 + lane];

  // ---- prefetch t = 0 operands
  v2f af[4];   // A-slices of F_t; also B-slices of F_t^T (identical layout)
  v8f q;       // Q_t in C-layout
  float u_;
#pragma unroll
  for (int kk = 0; kk < 4; ++kk)
    af[kk] = *(const v2f*)(Fb + lo * 16 + kk * 4 + 2 * hi);
#pragma unroll
  for (int i = 0; i < 8; ++i)
    q[i] = Qb[(i + 8 * hi) * 16 + lo];
  u_ = ub[lo];

  for (int t = 0; t < SEQ_; ++t) {
    // ---- prefetch t+1 (off critical path)
    const int tn = (t + 1 < SEQ_) ? t + 1 : t;
    v2f afn[4]; v8f qn; float un;
#pragma unroll
    for (int kk = 0; kk < 4; ++kk)
      afn[kk] = *(const v2f*)(Fb + (size_t)tn * 256 + lo * 16 + kk * 4 + 2 * hi);
#pragma unroll
    for (int i = 0; i < 8; ++i)
      qn[i] = Qb[(size_t)tn * 256 + (i + 8 * hi) * 16 + lo];
    un = ub[(size_t)tn * 16 + lo];

    // ---- x_t = F_t x_{t-1} + u_t : per-lane half-row partial + cross-half add
    float xp = 0.f;
#pragma unroll
    for (int kk = 0; kk < 4; ++kk) {
      const int c = kk * 4 + 2 * hi;
      xp += af[kk].x * xl[c] + af[kk].y * xl[c + 1];
    }
    xp += __shfl_xor(xp, 16);
    const float xnew = xp + u_;

    // ---- stage 1: T = F_t * P_{t-1}  (two independent 2-chains of WMMA)
    v8f z = {};
    v8f t01 = z, t23 = z;
#pragma unroll
    for (int kk = 0; kk < 4; ++kk) {
      const int r = kk * 4 + 2 * hi;                 // B rows r, r+1 per lane half
      v2f bp; bp.x = Pl[r * LP + lo]; bp.y = Pl[(r + 1) * LP + lo];
      if (kk < 2) t01 = wmma4(af[kk], bp, t01);
      else        t23 = wmma4(af[kk], bp, t23);
    }
#pragma unroll
    for (int i = 0; i < 8; ++i)
      Tl[(i + 8 * hi) * LP + lo] = t01[i] + t23[i];

    // ---- publish x state (xl reads for this step already done)
    if (lane < 16) {
      xl[lane] = xnew;
      xob[(size_t)t * 16 + lane] = xnew;
    }

    // ---- stage 2: P_t = T * F_t^T + Q_t  (B of F^T == A-regs of F)
    v8f p01 = q, p23 = z;
#pragma unroll
    for (int kk = 0; kk < 4; ++kk) {
      const v2f aT = *(const v2f*)&Tl[lo * LP + kk * 4 + 2 * hi];
      if (kk < 2) p01 = wmma4(aT, af[kk], p01);
      else        p23 = wmma4(aT, af[kk], p23);
    }
#pragma unroll
    for (int i = 0; i < 8; ++i) {
      const float pv = p01[i] + p23[i];
      Pl[(i + 8 * hi) * LP + lo] = pv;                       // next step's B source
      Pob[(size_t)t * 256 + (i + 8 * hi) * 16 + lo] = pv;    // coalesced output
    }

    // rotate prefetch registers
#pragma unroll
    for (int kk = 0; kk < 4; ++kk) af[kk] = afn[kk];
    q = qn; u_ = un;
  }
}

// ---------------------------------------------------------------------------
// Kernel 2: one wave32 per (b,t) tile; fully parallel, memory-bound.
// S~ = Hpad * P * Hpad^T + Rpad with Hpad = [H;0] (16x16); store 8x8 corner.
// ---------------------------------------------------------------------------
__global__ __launch_bounds__(256, 1) void lgssm_emit(
    const float* __restrict__ Hs, const float* __restrict__ Rs,
    const float* __restrict__ ds, const float* __restrict__ xs,
    const float* __restrict__ Ps,
    float* __restrict__ ys, float* __restrict__ Ss)
{
  __shared__ float Vl[8][XD_ * LP];
  __shared__ float xsh[8][XD_];

  const int w    = threadIdx.x >> 5;
  const int lane = threadIdx.x & 31;
  const int lo   = lane & 15;
  const int hi   = lane >> 4;

  const size_t tile = (size_t)blockIdx.x * 8 + w;   // flat b*SEQ + t
  const float* __restrict__ Ht = Hs + tile * 128;
  const float* __restrict__ Rt = Rs + tile * 64;
  const float* __restrict__ Pt = Ps + tile * 256;
  const float* __restrict__ xt = xs + tile * 16;
  const float* __restrict__ dt = ds + tile * 8;

  // A-slices of zero-padded H (rows 8..15 = 0)
  v2f ah[4];
#pragma unroll
  for (int kk = 0; kk < 4; ++kk) {
    v2f v = {0.f, 0.f};
    if (lo < 8) v = *(const v2f*)(Ht + lo * 16 + kk * 4 + 2 * hi);
    ah[kk] = v;
  }

  // stage 1: V = Hpad * P  (B-slices of P coalesced straight from global)
  v8f z = {};
  v8f v01 = z, v23 = z;
#pragma unroll
  for (int kk = 0; kk < 4; ++kk) {
    const int r = kk * 4 + 2 * hi;
    v2f bp; bp.x = Pt[r * 16 + lo]; bp.y = Pt[(r + 1) * 16 + lo];
    if (kk < 2) v01 = wmma4(ah[kk], bp, v01);
    else        v23 = wmma4(ah[kk], bp, v23);
  }
#pragma unroll
  for (int i = 0; i < 8; ++i)
    Vl[w][(i + 8 * hi) * LP + lo] = v01[i] + v23[i];

  // C operand = Rpad (top-left 8x8 = R, rest 0)
  v8f cr;
#pragma unroll
  for (int i = 0; i < 8; ++i) {
    const int m = i + 8 * hi;
    cr[i] = (m < 8 && lo < 8) ? Rt[m * 8 + lo] : 0.f;
  }

  // stage 2: S~ = V * Hpad^T + Rpad  (B of Hpad^T == A-regs of Hpad)
  v8f s01 = cr, s23 = z;
#pragma unroll
  for (int kk = 0; kk < 4; ++kk) {
    const v2f aV = *(const v2f*)&Vl[w][lo * LP + kk * 4 + 2 * hi];
    if (kk < 2) s01 = wmma4(aV, ah[kk], s01);
    else        s23 = wmma4(aV, ah[kk], s23);
  }
  if (hi == 0 && lo < 8) {
#pragma unroll
    for (int i = 0; i < 8; ++i)
      Ss[tile * 64 + i * 8 + lo] = s01[i] + s23[i];   // D vgpr i, lanes 0..7 = S[i][n]
  }

  // y = H x + d : split dot (8 cols per lane) + cross reduce
  if (lane < 16) xsh[w][lane] = xt[lane];
  const int m  = lo & 7;
  const int ch = lo >> 3;
  float yp = 0.f;
#pragma unroll
  for (int j = 0; j < 8; ++j)
    yp += Ht[m * 16 + ch * 8 + j] * xsh[w][ch * 8 + j];
  yp += __shfl_xor(yp, 8);
  if (hi == 0 && lo < 8)
    ys[tile * 8 + lo] = yp + dt[lo];
}

// ---------------------------------------------------------------------------
extern "C" void kernel_launch(void* const* d_in, const int* in_sizes, int n_in,
                              void* d_out, int out_size, void* d_ws, size_t ws_size,
                              hipStream_t stream) {
  const float* x0 = (const float*)d_in[0];
  const float* P0 = (const float*)d_in[1];
  const float* Fs = (const float*)d_in[2];
  const float* Hs = (const float*)d_in[3];
  const float* Qs = (const float*)d_in[4];
  const float* Rs = (const float*)d_in[5];
  const float* us = (const float*)d_in[6];
  const float* ds = (const float*)d_in[7];

  float* out = (float*)d_out;
  float* xs = out;                                        // (64,2048,16)
  float* Ps = xs + (size_t)BS_ * SEQ_ * XD_;              // (64,2048,16,16)
  float* ys = Ps + (size_t)BS_ * SEQ_ * XD_ * XD_;        // (64,2048,8)
  float* Ss = ys + (size_t)BS_ * SEQ_ * YD_;              // (64,2048,8,8)

  lgssm_scan<<<BS_, 32, 0, stream>>>(x0, P0, Fs, Qs, us, xs, Ps);
  lgssm_emit<<<(BS_ * SEQ_) / 8, 256, 0, stream>>>(Hs, Rs, ds, xs, Ps, ys, Ss);
}